// HeteroGraphTransformerLayer_89936615178983
// MI455X (gfx1250) — compile-verified
//
#include <hip/hip_runtime.h>
#include <math.h>

// ---------------- problem constants (match reference) ----------------
#define NHEAD 4
#define DHEAD 32
#define HID   128
#define EDIM  32
#define FFD   256
#define NNODE 50000
#define NEDGE 400000
#define INV_SCALE 0.17677669529663687f   // 1/sqrt(32)

typedef __attribute__((ext_vector_type(2))) float v2f;
typedef __attribute__((ext_vector_type(8))) float v8f;

// D = A(16x4,f32) * B(4x16,f32) + C(16x16,f32)   -- CDNA5 fp32 WMMA
__device__ __forceinline__ v8f wmma4(v2f a, v2f b, v8f c) {
  return __builtin_amdgcn_wmma_f32_16x16x4_f32(false, a, false, b, (short)0, c,
                                               false, false);
}

// ---- 16x128 strip GEMM core, A from global --------------------------------
// Arow = A + (r0+m)*K + 2*half ; Bp = B + (2*half)*ldb + n(+group offset)
// Fully unrolled; per k-step: 1 b64 A-load + 16 b32 B-loads into fresh regs,
// then 8 independent WMMAs -> scheduler can overlap next-step loads.
template <int K>
__device__ __forceinline__ void strip_gemm_g(const float* __restrict__ Arow,
                                             const float* __restrict__ Bp,
                                             const int ldb, v8f* acc) {
#pragma unroll
  for (int k = 0; k < K; k += 4) {
    v2f a = *reinterpret_cast<const v2f*>(Arow + k);
    const float* Bk = Bp + (size_t)k * ldb;
    v2f bf[8];
#pragma unroll
    for (int nt = 0; nt < 8; ++nt) {
      bf[nt].x = Bk[nt * 16];
      bf[nt].y = Bk[ldb + nt * 16];
    }
#pragma unroll
    for (int nt = 0; nt < 8; ++nt) acc[nt] = wmma4(a, bf[nt], acc[nt]);
  }
}

// ---- 16x128 strip GEMM core, A from LDS ([feat][node], stride 17) ---------
// Al = ldsA + m ; Bp = B + (2*half)*ldb + n(+group offset)
template <int K>
__device__ __forceinline__ void strip_gemm_l(const float* __restrict__ Al,
                                             const float* __restrict__ Bp,
                                             const int ldb, const int half, v8f* acc) {
#pragma unroll
  for (int k = 0; k < K; k += 4) {
    v2f a;
    a.x = Al[(k + 2 * half) * 17];
    a.y = Al[(k + 2 * half + 1) * 17];
    const float* Bk = Bp + (size_t)k * ldb;
    v2f bf[8];
#pragma unroll
    for (int nt = 0; nt < 8; ++nt) {
      bf[nt].x = Bk[nt * 16];
      bf[nt].y = Bk[ldb + nt * 16];
    }
#pragma unroll
    for (int nt = 0; nt < 8; ++nt) acc[nt] = wmma4(a, bf[nt], acc[nt]);
  }
}

__device__ __forceinline__ void atomicMaxF(float* addr, float val) {
  unsigned int* ua = (unsigned int*)addr;
  unsigned int cur = __float_as_uint(*addr);
  while (__uint_as_float(cur) < val) {
    unsigned int prev = atomicCAS(ua, cur, __float_as_uint(val));
    if (prev == cur) break;
    cur = prev;
  }
}

// ---------------- projection GEMM: C[Mx128] = A[Mx128] @ B[128x128] --------
__global__ void gemm_proj(const float* __restrict__ A, const float* __restrict__ B,
                          float* __restrict__ C, int M) {
  const int wave = threadIdx.x >> 5;
  const int lane = threadIdx.x & 31;
  const int rt = blockIdx.x * (blockDim.x >> 5) + wave;
  if (rt >= (M >> 4)) return;                 // whole-wave exit
  const int r0 = rt << 4;
  const int m = lane & 15, half = lane >> 4, n = m;

  v8f acc[8] = {};
  strip_gemm_g<HID>(A + (size_t)(r0 + m) * HID + 2 * half,
                    B + (size_t)(2 * half) * HID + n, HID, acc);
#pragma unroll
  for (int nt = 0; nt < 8; ++nt)
    for (int i = 0; i < 8; ++i)
      C[(size_t)(r0 + i + 8 * half) * HID + nt * 16 + n] = acc[nt][i];
}

// ---------------- segment-accumulator init --------------------------------
__global__ void init_seg(float* __restrict__ mbuf, float* __restrict__ den,
                         float* __restrict__ num, int nNodes) {
  int i = blockIdx.x * blockDim.x + threadIdx.x;
  if (i < nNodes * NHEAD) { mbuf[i] = -1e30f; den[i] = 0.f; }
  if (i < nNodes * HID)   num[i] = 0.f;
}

// ---------------- edge passes ----------------------------------------------
// Each wave handles 16 edges. Ep = ea[e0:e0+16] @ we computed by WMMA into
// padded LDS ([feat][edge], stride 17), then 64 (edge,head) units / 32 lanes.
#define EDGE_WAVES 4
__global__ void edge_pass(const float* __restrict__ Q, const float* __restrict__ Kp,
                          const float* __restrict__ Vp,
                          const float* __restrict__ ea, const float* __restrict__ we,
                          const int* __restrict__ src, const int* __restrict__ dst,
                          float* __restrict__ mbuf, float* __restrict__ den,
                          float* __restrict__ num, int pass) {
  __shared__ float eplds[EDGE_WAVES][HID * 17];
  const int wave = threadIdx.x >> 5;
  const int lane = threadIdx.x & 31;
  const int e0 = (blockIdx.x * EDGE_WAVES + wave) << 4;
  float* epl = eplds[wave];

  // --- Ep tile via WMMA: A = ea rows (16x32 contiguous), B = we (32x128) ---
  {
    const int m = lane & 15, half = lane >> 4, n = m;
    v8f acc[8] = {};
    strip_gemm_g<EDIM>(ea + (size_t)(e0 + m) * EDIM + 2 * half,
                       we + (size_t)(2 * half) * HID + n, HID, acc);
#pragma unroll
    for (int nt = 0; nt < 8; ++nt)
      for (int i = 0; i < 8; ++i)
        epl[(nt * 16 + n) * 17 + i + 8 * half] = acc[nt][i];   // [feat][edge]
  }
  // same-wave producer/consumer; DS ops are in order within a wave.

  // --- per (edge, head) processing: 64 units, 2 per lane ---
  for (int iter = 0; iter < 2; ++iter) {
    const int h = (lane >> 4) + 2 * iter;   // head 0..3
    const int e = lane & 15;                // local edge
    const int s  = src[e0 + e];
    const int dt = dst[e0 + e];
    const float4* qv = (const float4*)(Q  + (size_t)dt * HID + h * DHEAD);
    const float4* kv = (const float4*)(Kp + (size_t)s  * HID + h * DHEAD);
    float dot = 0.f;
    for (int j = 0; j < 8; ++j) {
      float4 q4 = qv[j], k4 = kv[j];
      const int f = h * DHEAD + j * 4;
      dot += q4.x * (k4.x + epl[(f + 0) * 17 + e]);
      dot += q4.y * (k4.y + epl[(f + 1) * 17 + e]);
      dot += q4.z * (k4.z + epl[(f + 2) * 17 + e]);
      dot += q4.w * (k4.w + epl[(f + 3) * 17 + e]);
    }
    const float logit = dot * INV_SCALE;
    if (pass == 0) {
      atomicMaxF(&mbuf[dt * NHEAD + h], logit);
    } else {
      const float ex = expf(logit - mbuf[dt * NHEAD + h]);
      atomicAdd(&den[dt * NHEAD + h], ex);
      const float4* vv = (const float4*)(Vp + (size_t)s * HID + h * DHEAD);
      for (int j = 0; j < 8; ++j) {
        float4 v4 = vv[j];
        const int f = h * DHEAD + j * 4;
        atomicAdd(&num[(size_t)dt * HID + f + 0], ex * (v4.x + epl[(f + 0) * 17 + e]));
        atomicAdd(&num[(size_t)dt * HID + f + 1], ex * (v4.y + epl[(f + 1) * 17 + e]));
        atomicAdd(&num[(size_t)dt * HID + f + 2], ex * (v4.z + epl[(f + 2) * 17 + e]));
        atomicAdd(&num[(size_t)dt * HID + f + 3], ex * (v4.w + epl[(f + 3) * 17 + e]));
      }
    }
  }
}

// ---------------- fused node update ----------------------------------------
// one wave = 16 nodes: skip GEMM + msg + LN1 + FF1(gelu) + FF2 + LN2 + store.
__device__ __forceinline__ void ln_inplace(float* buf, const float* __restrict__ g,
                                           const float* __restrict__ b, int lane) {
  const int node = lane & 15, part = lane >> 4;
  float s = 0.f, ss = 0.f;
  for (int f = part * 64; f < part * 64 + 64; ++f) {
    float v = buf[f * 17 + node]; s += v; ss += v * v;
  }
  s  += __shfl_xor(s, 16, 32);
  ss += __shfl_xor(ss, 16, 32);
  const float mu  = s * (1.f / 128.f);
  const float var = ss * (1.f / 128.f) - mu * mu;
  const float inv = 1.f / sqrtf(var + 1e-5f);
  for (int f = part * 64; f < part * 64 + 64; ++f)
    buf[f * 17 + node] = (buf[f * 17 + node] - mu) * inv * g[f] + b[f];
}

#define NU_WAVES 2
__global__ void node_update(const float* __restrict__ h,
                            const float* __restrict__ num, const float* __restrict__ den,
                            const float* __restrict__ wskip,
                            const float* __restrict__ g1, const float* __restrict__ b1,
                            const float* __restrict__ g2, const float* __restrict__ b2,
                            const float* __restrict__ wf1, const float* __restrict__ bf1,
                            const float* __restrict__ wf2, const float* __restrict__ bf2,
                            float* __restrict__ out, int nNodes) {
  __shared__ float b1s[NU_WAVES][HID * 17];
  __shared__ float b2s[NU_WAVES][FFD * 17];
  const int wave = threadIdx.x >> 5;
  const int lane = threadIdx.x & 31;
  const int tile = blockIdx.x * NU_WAVES + wave;
  if (tile >= (nNodes >> 4)) return;           // whole-wave exit, WMMA safe
  const int r0 = tile << 4;
  float* buf1 = b1s[wave];
  float* buf2 = b2s[wave];
  const int m = lane & 15, half = lane >> 4, n = m;

  // ---- skip = h@wskip ; buf1 = h + skip + msg ----
  {
    v8f acc[8] = {};
    strip_gemm_g<HID>(h + (size_t)(r0 + m) * HID + 2 * half,
                      wskip + (size_t)(2 * half) * HID + n, HID, acc);
#pragma unroll
    for (int nt = 0; nt < 8; ++nt) {
      const int ng = nt * 16 + n;
      for (int i = 0; i < 8; ++i) {
        const int mr = i + 8 * half;
        const size_t row = (size_t)(r0 + mr);
        const float msg = num[row * HID + ng] / (den[row * NHEAD + (ng >> 5)] + 1e-8f);
        buf1[ng * 17 + mr] = acc[nt][i] + h[row * HID + ng] + msg;
      }
    }
  }
  ln_inplace(buf1, g1, b1, lane);

  // ---- FF1: buf2 = gelu(buf1 @ wf1 + bf1), 2 groups of 8 column tiles ----
  for (int g = 0; g < 2; ++g) {
    v8f acc[8] = {};
    strip_gemm_l<HID>(buf1 + m,
                      wf1 + (size_t)(2 * half) * FFD + n + g * 128, FFD, half, acc);
#pragma unroll
    for (int nt = 0; nt < 8; ++nt) {
      const int ng = g * 128 + nt * 16 + n;
      const float bb = bf1[ng];
      for (int i = 0; i < 8; ++i) {
        const float x = acc[nt][i] + bb;
        buf2[ng * 17 + i + 8 * half] = 0.5f * x * (1.f + erff(x * 0.70710678118654752f));
      }
    }
  }

  // ---- FF2: buf1 = buf1 + (buf2 @ wf2 + bf2) ----
  {
    v8f acc[8] = {};
    strip_gemm_l<FFD>(buf2 + m,
                      wf2 + (size_t)(2 * half) * HID + n, HID, half, acc);
#pragma unroll
    for (int nt = 0; nt < 8; ++nt) {
      const int ng = nt * 16 + n;
      const float bb = bf2[ng];
      for (int i = 0; i < 8; ++i) {
        const int mr = i + 8 * half;
        buf1[ng * 17 + mr] += acc[nt][i] + bb;
      }
    }
  }

  // ---- LN2 + writeout ----
  {
    const int node = lane & 15, part = lane >> 4;
    float s = 0.f, ss = 0.f;
    for (int f = part * 64; f < part * 64 + 64; ++f) {
      float v = buf1[f * 17 + node]; s += v; ss += v * v;
    }
    s  += __shfl_xor(s, 16, 32);
    ss += __shfl_xor(ss, 16, 32);
    const float mu  = s * (1.f / 128.f);
    const float var = ss * (1.f / 128.f) - mu * mu;
    const float inv = 1.f / sqrtf(var + 1e-5f);
    for (int f = part * 64; f < part * 64 + 64; ++f)
      out[(size_t)(r0 + node) * HID + f] =
          (buf1[f * 17 + node] - mu) * inv * g2[f] + b2[f];
  }
}

// ---------------- host-side orchestration ----------------------------------
extern "C" void kernel_launch(void* const* d_in, const int* in_sizes, int n_in,
                              void* d_out, int out_size, void* d_ws, size_t ws_size,
                              hipStream_t stream) {
  const float* x_a    = (const float*)d_in[0];
  const float* x_b    = (const float*)d_in[1];
  const float* ea_ab  = (const float*)d_in[2];
  const float* ea_ba  = (const float*)d_in[3];
  const int*   ei_ab  = (const int*)  d_in[4];
  const int*   ei_ba  = (const int*)  d_in[5];
  const float* wq_ab  = (const float*)d_in[6];
  const float* wk_ab  = (const float*)d_in[7];
  const float* we_ab  = (const float*)d_in[8];
  const float* wv_ab  = (const float*)d_in[9];
  const float* wq_ba  = (const float*)d_in[10];
  const float* wk_ba  = (const float*)d_in[11];
  const float* we_ba  = (const float*)d_in[12];
  const float* wv_ba  = (const float*)d_in[13];
  const float* wskip_a = (const float*)d_in[14];
  const float* wskip_b = (const float*)d_in[15];
  const float* wf1_a = (const float*)d_in[16];
  const float* bf1_a = (const float*)d_in[17];
  const float* wf2_a = (const float*)d_in[18];
  const float* bf2_a = (const float*)d_in[19];
  const float* g1_a  = (const float*)d_in[20];
  const float* b1_a  = (const float*)d_in[21];
  const float* g2_a  = (const float*)d_in[22];
  const float* b2_a  = (const float*)d_in[23];
  const float* wf1_b = (const float*)d_in[24];
  const float* bf1_b = (const float*)d_in[25];
  const float* wf2_b = (const float*)d_in[26];
  const float* bf2_b = (const float*)d_in[27];
  const float* g1_b  = (const float*)d_in[28];
  const float* b1_b  = (const float*)d_in[29];
  const float* g2_b  = (const float*)d_in[30];
  const float* b2_b  = (const float*)d_in[31];

  float* out_a = (float*)d_out;
  float* out_b = out_a + (size_t)NNODE * HID;

  // workspace layout (floats); directions run sequentially so Q/K/V reused
  const size_t SZ = (size_t)NNODE * HID;           // 6.4e6 floats
  float* ws    = (float*)d_ws;
  float* Qb    = ws;
  float* Kb    = ws + SZ;
  float* Vb    = ws + 2 * SZ;
  float* num_b = ws + 3 * SZ;
  float* num_a = ws + 4 * SZ;
  float* m_b   = ws + 5 * SZ;
  float* den_b = m_b + (size_t)NNODE * NHEAD;
  float* m_a   = den_b + (size_t)NNODE * NHEAD;
  float* den_a = m_a + (size_t)NNODE * NHEAD;

  const int rowStrips = NNODE / 16;                       // 3125
  const int gemmGrid  = (rowStrips + 3) / 4;              // 4 waves/block
  const int edgeGrid  = NEDGE / (16 * EDGE_WAVES);        // 6250
  const int initGrid  = (NNODE * HID + 255) / 256;
  const int nuGrid    = (rowStrips + NU_WAVES - 1) / NU_WAVES;

  // ---------- direction a->b : messages land on b ----------
  gemm_proj<<<gemmGrid, 128, 0, stream>>>(x_b, wq_ab, Qb, NNODE);
  gemm_proj<<<gemmGrid, 128, 0, stream>>>(x_a, wk_ab, Kb, NNODE);
  gemm_proj<<<gemmGrid, 128, 0, stream>>>(x_a, wv_ab, Vb, NNODE);
  init_seg<<<initGrid, 256, 0, stream>>>(m_b, den_b, num_b, NNODE);
  edge_pass<<<edgeGrid, 128, 0, stream>>>(Qb, Kb, Vb, ea_ab, we_ab,
                                          ei_ab, ei_ab + NEDGE, m_b, den_b, num_b, 0);
  edge_pass<<<edgeGrid, 128, 0, stream>>>(Qb, Kb, Vb, ea_ab, we_ab,
                                          ei_ab, ei_ab + NEDGE, m_b, den_b, num_b, 1);

  // ---------- direction b->a : messages land on a ----------
  gemm_proj<<<gemmGrid, 128, 0, stream>>>(x_a, wq_ba, Qb, NNODE);
  gemm_proj<<<gemmGrid, 128, 0, stream>>>(x_b, wk_ba, Kb, NNODE);
  gemm_proj<<<gemmGrid, 128, 0, stream>>>(x_b, wv_ba, Vb, NNODE);
  init_seg<<<initGrid, 256, 0, stream>>>(m_a, den_a, num_a, NNODE);
  edge_pass<<<edgeGrid, 128, 0, stream>>>(Qb, Kb, Vb, ea_ba, we_ba,
                                          ei_ba, ei_ba + NEDGE, m_a, den_a, num_a, 0);
  edge_pass<<<edgeGrid, 128, 0, stream>>>(Qb, Kb, Vb, ea_ba, we_ba,
                                          ei_ba, ei_ba + NEDGE, m_a, den_a, num_a, 1);

  // ---------- node updates ----------
  node_update<<<nuGrid, 64, 0, stream>>>(x_a, num_a, den_a, wskip_a,
                                         g1_a, b1_a, g2_a, b2_a,
                                         wf1_a, bf1_a, wf2_a, bf2_a, out_a, NNODE);
  node_update<<<nuGrid, 64, 0, stream>>>(x_b, num_b, den_b, wskip_b,
                                         g1_b, b1_b, g2_b, b2_b,
                                         wf1_b, bf1_b, wf2_b, bf2_b, out_b, NNODE);
}